// GreedyGraphTransformerBaseline_35931696399098
// MI455X (gfx1250) — compile-verified
//
#include <hip/hip_runtime.h>
#include <math.h>
#include <stdint.h>

// Problem constants (match reference)
#define BB 2048
#define NN 200
#define HH 128
#define TT 180
#define PSTR 130   // padded LDS stride for P (kills 64-bank conflicts at stride 128)
#define WSTR 130   // padded LDS stride for W
#define MT 13      // 16-row tiles covering 208 (>=200) rows

typedef __attribute__((ext_vector_type(2))) float v2f;
typedef __attribute__((ext_vector_type(8))) float v8f;

static __device__ inline v8f wmma_f32_4(v2f a, v2f b, v8f c) {
  // verified: lowers to v_wmma_f32_16x16x4_f32 v[d:d+7], v[a:a+1], v[b:b+1], v[c..]
  return __builtin_amdgcn_wmma_f32_16x16x4_f32(false, a, false, b,
                                               (short)0, c, false, false);
}

__global__ __launch_bounds__(256, 1)
void vrp_fused(const float* __restrict__ x, const float* __restrict__ Wm,
               const float* __restrict__ bias, const float* __restrict__ coords,
               const float* __restrict__ dems, const float* __restrict__ caps,
               float* __restrict__ out)
{
  // ~265 KB static LDS (<= 320 KB/WGP on CDNA5)
  __shared__ float sG[NN * NN];        // 160,000 B; W aliased here during phase 1
  __shared__ float sP[208 * PSTR];     // 108,160 B, padded stride
  __shared__ float sCX[208], sCY[208], sDem[208], sBias[HH];

  const int b    = blockIdx.x;
  const int tid  = threadIdx.x;
  const int wave = tid >> 5;           // 8 waves of 32 (wave32)
  const int lane = tid & 31;
  const int ll   = lane & 15;
  const int hs   = lane >> 4;          // 0: holds K+0..1, 1: holds K+2..3

  float* sW = sG;                      // alias: W lives in G region during phase 1

  // log_probs are identically zero
  for (int i = tid; i < TT; i += 256)
    out[(size_t)BB * TT + (size_t)b * TT + i] = 0.0f;

  // ---- Stage W into LDS via CDNA5 async global->LDS DMA (b64: 520B row
  // stride is 8B-aligned only). 8192 b64 transfers, 32 per thread.
  {
    const unsigned ldsW = (unsigned)(uintptr_t)(&sW[0]);   // wave-relative LDS offset
    const uint64_t gW   = (uint64_t)(uintptr_t)Wm;
    for (int i = tid; i < (HH * HH) / 2; i += 256) {       // i indexes float pairs
      const unsigned dst = ldsW + (unsigned)((i >> 6) * (WSTR * 4) + (i & 63) * 8);
      const uint64_t src = gW + (uint64_t)i * 8u;
      asm volatile("global_load_async_to_lds_b64 %0, %1, off"
                   :: "v"(dst), "v"(src) : "memory");
    }
    asm volatile("s_wait_asynccnt 0x0" ::: "memory");
  }

  // Stage bias, coords, demands
  for (int i = tid; i < HH; i += 256) sBias[i] = bias[i];
  for (int i = tid; i < 208; i += 256) {
    const int j = (i < NN) ? i : (NN - 1);
    sCX[i]  = coords[((size_t)b * NN + j) * 2 + 0];
    sCY[i]  = coords[((size_t)b * NN + j) * 2 + 1];
    sDem[i] = dems[(size_t)b * NN + j];
  }
  __syncthreads();

  // ---- Phase 1: P = x[b] @ W^T + bias  (M=200(208), O=128, K=128), fp32 WMMA.
  // A row-chunk cached in 64 VGPRs -> x is streamed from HBM exactly once.
  const float* xb = x + (size_t)b * NN * HH;
  for (int mt = wave; mt < MT; mt += 8) {
    const int m  = mt * 16 + ll;
    const int mc = (m < NN) ? m : (NN - 1);      // clamp pad rows (discarded anyway)
    const float* ap = xb + (size_t)mc * HH + hs * 2;
    v2f a[32];
    #pragma unroll
    for (int k = 0; k < 32; ++k) a[k] = *(const v2f*)(ap + 4 * k);
    const int rbase = mt * 16 + hs * 8;
    for (int ot = 0; ot < 8; ++ot) {
      const int o = ot * 16 + ll;
      const float* bp = &sW[o * WSTR + hs * 2];  // B[k][o] = W[o][k]
      v8f c = {};
      #pragma unroll
      for (int k = 0; k < 32; ++k) {
        const v2f bv = *(const v2f*)(bp + 4 * k);
        c = wmma_f32_4(a[k], bv, c);
      }
      const float bo = sBias[o];
      #pragma unroll
      for (int g = 0; g < 8; ++g)
        sP[(rbase + g) * PSTR + o] = c[g] + bo;  // rbase+g <= 207, always in sP
    }
  }
  __syncthreads();

  // ---- Phase 2: G = P @ P^T  (200x200, K=128), fp32 WMMA, A cached per mt.
  for (int mt = wave; mt < MT; mt += 8) {
    const float* ap = &sP[(mt * 16 + ll) * PSTR + hs * 2];
    v2f a[32];
    #pragma unroll
    for (int k = 0; k < 32; ++k) a[k] = *(const v2f*)(ap + 4 * k);
    const int rbase = mt * 16 + hs * 8;
    for (int nt = 0; nt < MT; ++nt) {
      const float* bp = &sP[(nt * 16 + ll) * PSTR + hs * 2];  // B[k][n] = P[n][k]
      v8f c = {};
      #pragma unroll
      for (int k = 0; k < 32; ++k) {
        const v2f bv = *(const v2f*)(bp + 4 * k);
        c = wmma_f32_4(a[k], bv, c);
      }
      const int col = nt * 16 + ll;
      if (col < NN) {
        #pragma unroll
        for (int g = 0; g < 8; ++g) {
          const int r = rbase + g;
          if (r < NN) sG[r * NN + col] = c[g];
        }
      }
    }
  }
  __syncthreads();

  // ---- Phase 3: 180 sequential greedy steps, single wave, all data in LDS.
  // Lane L owns nodes {L, L+32, ...}; visited kept as a lane-local bitmask.
  if (wave == 0) {
    const float cap0 = caps[b];
    float rem = cap0;
    int cur = 0;
    unsigned visited = 0;
    float* acts = out + (size_t)b * TT;
    for (int step = 0; step < TT; ++step) {
      const float cx = sCX[cur], cy = sCY[cur];
      const float* grow = &sG[cur * NN];         // sims row: (E E^T)[cur,:]
      float bv = -INFINITY; int bi = 0x7fffffff;
      #pragma unroll
      for (int s = 0; s < 7; ++s) {
        const int j = lane + s * 32;
        if (j < NN) {
          const float dx = sCX[j] - cx, dy = sCY[j] - cy;
          const float sc = grow[j] - 0.1f * sqrtf(dx * dx + dy * dy);
          bool masked = (((visited >> s) & 1u) != 0u) || (sDem[j] > rem);
          if (j == 0) masked = false;            // depot always allowed
          if (!masked && (sc > bv || (sc == bv && j < bi))) { bv = sc; bi = j; }
        }
      }
      // wave32 argmax, first-occurrence tie-break (matches jnp.argmax)
      #pragma unroll
      for (int off = 16; off >= 1; off >>= 1) {
        const float ov = __shfl_xor(bv, off, 32);
        const int   oi = __shfl_xor(bi, off, 32);
        if (ov > bv || (ov == bv && oi < bi)) { bv = ov; bi = oi; }
      }
      const int nxt = bi;
      if ((nxt & 31) == lane) visited |= (1u << (nxt >> 5));
      rem = (nxt == 0) ? cap0 : (rem - sDem[nxt]);
      cur = nxt;
      if (lane == 0) acts[step] = (float)nxt;    // actions emitted as float indices
    }
  }
}

extern "C" void kernel_launch(void* const* d_in, const int* in_sizes, int n_in,
                              void* d_out, int out_size, void* d_ws, size_t ws_size,
                              hipStream_t stream) {
  const float* x   = (const float*)d_in[0];
  const float* W   = (const float*)d_in[1];
  const float* bia = (const float*)d_in[2];
  const float* co  = (const float*)d_in[3];
  const float* de  = (const float*)d_in[4];
  const float* ca  = (const float*)d_in[5];
  (void)in_sizes; (void)n_in; (void)d_ws; (void)ws_size; (void)out_size;
  vrp_fused<<<BB, 256, 0, stream>>>(x, W, bia, co, de, ca, (float*)d_out);
}